// KGEModel_6786048327924
// MI455X (gfx1250) — compile-verified
//
#include <hip/hip_runtime.h>
#include <hip/hip_bf16.h>

// RotatE scoring: score[b,n] = GAMMA - sum_d sqrt(|rot_head[b,d] - tail[n,d]|^2)
// Memory-bound gather over the entity table; tail rows are streamed into LDS
// via CDNA5 async global->LDS copies (ASYNCcnt) with per-wave double buffering.

#define GAS __attribute__((address_space(1)))
#define LAS __attribute__((address_space(3)))

#if defined(__HIP_DEVICE_COMPILE__) && defined(__gfx1250__) && \
    __has_builtin(__builtin_amdgcn_global_load_async_to_lds_b128)
#define ROTATE_ASYNC_LDS 1
#else
#define ROTATE_ASYNC_LDS 0
#endif

typedef int v4i_t __attribute__((ext_vector_type(4)));

namespace {
constexpr int   kDim       = 1000;  // entity embedding dim (2 * 500)
constexpr int   kHalf      = 500;   // complex dims / relation dim
constexpr int   kWaves     = 4;     // waves per block
constexpr int   kThreads   = kWaves * 32;
constexpr int   kNegPerBlk = 16;    // negatives handled per block
constexpr float kGamma     = 12.0f;
// phase = rel / (EMB_RANGE/PI), EMB_RANGE = (12+2)/500
constexpr float kPhaseScale = 3.14159265358979323846f * 500.0f / 14.0f;
}  // namespace

template <int N>
__device__ __forceinline__ void wait_async() {
#if ROTATE_ASYNC_LDS
#if __has_builtin(__builtin_amdgcn_s_wait_asynccnt)
  __builtin_amdgcn_s_wait_asynccnt(N);
#else
  asm volatile("s_wait_asynccnt %0" ::"n"(N) : "memory");
#endif
#endif
}

// Copy one 4000-byte entity row (re[0..500) || im[0..500)) into LDS.
// Async path: 8 x global_load_async_to_lds_b128 per wave (7 full + 26 lanes).
__device__ __forceinline__ void copy_row_to_lds(const float* __restrict__ gsrc,
                                                float* ldst, int lane) {
#if ROTATE_ASYNC_LDS
  GAS v4i_t* g = (GAS v4i_t*)gsrc;
  LAS v4i_t* l = (LAS v4i_t*)ldst;
#pragma unroll
  for (int i = 0; i < 7; ++i) {
    const int e = i * 32 + lane;
    __builtin_amdgcn_global_load_async_to_lds_b128(g + e, l + e, 0, 0);
  }
  if (lane < 26) {
    const int e = 224 + lane;
    __builtin_amdgcn_global_load_async_to_lds_b128(g + e, l + e, 0, 0);
  }
#else
  const float4* g4 = (const float4*)gsrc;
  float4*       l4 = (float4*)ldst;
#pragma unroll
  for (int i = 0; i < 7; ++i) l4[i * 32 + lane] = g4[i * 32 + lane];
  if (lane < 26) l4[224 + lane] = g4[224 + lane];
#endif
}

__global__ __launch_bounds__(kThreads) void rotate_score_kernel(
    const int* __restrict__ head_part, const int* __restrict__ tail_part,
    const float* __restrict__ rel_emb, const float* __restrict__ ent_emb,
    float* __restrict__ out, int B, int NEG) {
  __shared__ float sRot[kDim];                    // rotated head for this b
  __shared__ float sTail[kWaves * 2 * kDim];      // per-wave double buffers

  const int b       = blockIdx.x;
  const int negBase = blockIdx.y * kNegPerBlk;
  const int tid     = threadIdx.x;
  const int lane    = tid & 31;
  const int wv      = tid >> 5;

  // ---- Phase 1: rotate head embedding into LDS (depends only on b,d) ----
  {
    const int h = head_part[b * 3 + 0];
    const int r = head_part[b * 3 + 1];
    const float* __restrict__ hrow = ent_emb + (size_t)h * kDim;
    const float* __restrict__ rrow = rel_emb + (size_t)r * kHalf;
    for (int d = tid; d < kHalf; d += kThreads) {
      float s, c;
      sincosf(rrow[d] * kPhaseScale, &s, &c);  // precise: |phase| up to ~500 rad
      const float re = hrow[d];
      const float im = hrow[d + kHalf];
      sRot[d]         = re * c - im * s;       // re_head*re_rel - im_head*im_rel
      sRot[d + kHalf] = re * s + im * c;       // re_head*im_rel + im_head*re_rel
    }
  }
  __syncthreads();

  // ---- Phase 2: each wave streams 4 tail rows, double-buffered in LDS ----
  float* bufA = sTail + wv * (2 * kDim);
  float* bufB = bufA + kDim;
  constexpr int kIters = kNegPerBlk / kWaves;  // 4 negatives per wave

  const int n0 = negBase + wv;
  if (n0 < NEG) {
    const int t0 = tail_part[b * NEG + n0];
    copy_row_to_lds(ent_emb + (size_t)t0 * kDim, bufA, lane);
  }

#pragma unroll
  for (int i = 0; i < kIters; ++i) {
    const int neg = negBase + wv + i * kWaves;
    if (neg >= NEG) break;
    float* cur = (i & 1) ? bufB : bufA;
    float* nxt = (i & 1) ? bufA : bufB;

    const int  negNext = neg + kWaves;
    const bool hasNext = (i + 1 < kIters) && (negNext < NEG);
    if (hasNext) {
      const int tn = tail_part[b * NEG + negNext];
      copy_row_to_lds(ent_emb + (size_t)tn * kDim, nxt, lane);
      wait_async<8>();   // 8 newer async loads outstanding -> row i complete
    } else {
      wait_async<0>();
    }

    float acc = 0.0f;
#pragma unroll
    for (int k = 0; k < 16; ++k) {
      const int d = lane + (k << 5);
      if (d < kHalf) {
        const float rr = sRot[d]         - cur[d];
        const float ii = sRot[d + kHalf] - cur[d + kHalf];
        acc += sqrtf(rr * rr + ii * ii);
      }
    }
    // wave32 reduction
#pragma unroll
    for (int off = 16; off > 0; off >>= 1) acc += __shfl_down(acc, off, 32);
    if (lane == 0) out[(size_t)b * NEG + neg] = kGamma - acc;
  }
}

extern "C" void kernel_launch(void* const* d_in, const int* in_sizes, int n_in,
                              void* d_out, int out_size, void* d_ws, size_t ws_size,
                              hipStream_t stream) {
  (void)n_in; (void)d_ws; (void)ws_size; (void)out_size;
  const int*   head_part = (const int*)d_in[0];    // (B, 3) int32
  const int*   tail_part = (const int*)d_in[1];    // (B, NEG) int32
  const float* rel_emb   = (const float*)d_in[2];  // (N_REL, 500) f32
  const float* ent_emb   = (const float*)d_in[3];  // (N_ENT, 1000) f32
  float*       out       = (float*)d_out;          // (B, NEG) f32

  const int B   = in_sizes[0] / 3;
  const int NEG = (B > 0) ? in_sizes[1] / B : 0;
  if (B <= 0 || NEG <= 0) return;

  dim3 grid(B, (NEG + kNegPerBlk - 1) / kNegPerBlk);
  rotate_score_kernel<<<grid, kThreads, 0, stream>>>(head_part, tail_part,
                                                     rel_emb, ent_emb, out, B, NEG);
}